// CRF_37718402793986
// MI455X (gfx1250) — compile-verified
//
#include <hip/hip_runtime.h>

// ---------------------------------------------------------------------------
// CRF objective on MI455X (gfx1250, wave32).
//
// Separable-scan math:
//   out = mean_b [ C*( sum_j tp[b,0,j]
//                      + sum_{t=1..lens[b]} (32*m_{b,t} + sum_i log S_i) )
//                  - sum_h[b] - sum_g[b] ]
//   m = max_j e_j, p_j = exp(e_j - m), S_i = sum_j exp(trans[i,j]) * p_j.
//
// S = expT * p as WMMA f32 16x16x4: 16 (b,t) pairs = A rows, K=32 chained as
// 8 x (K=4), two B matrices for i=0..15 / 16..31 -> 16 v_wmma per 16 pairs.
// VALU trimmed: raw v_log (log2, inputs >= 1 so no denorm fixup needed),
// per-lane accumulation (block reduction at the end sums lanes), masked path
// only for partially-valid boundary tiles.
// ---------------------------------------------------------------------------

typedef float v2f __attribute__((ext_vector_type(2)));
typedef float v8f __attribute__((ext_vector_type(8)));

#if __has_builtin(__builtin_amdgcn_logf)
#define RAW_LOG2(x) __builtin_amdgcn_logf(x)   // raw v_log_f32 (log2)
#else
#define RAW_LOG2(x) __log2f(x)
#endif

#define B_  512
#define T_  2048
#define C_  32
#define CHUNKS_PER_B 4                       // t-chunks per batch row
#define CHUNK_T      (T_ / CHUNKS_PER_B)     // 512
#define WAVES        8
#define ITERS        4                       // 8 waves * 4 * 16 pairs = 512 t

#define LN2_F        0.69314718055994530942f

__global__ __launch_bounds__(256)
void crf_main(const float* __restrict__ tp,     // (B,T,C) f32
              const float* __restrict__ trans,  // (C,C)   f32
              const int*   __restrict__ tt,     // (B,T)   i32
              const int*   __restrict__ lens,   // (B,)    i32
              float* __restrict__ partials)     // (gridDim.x,)
{
  __shared__ float s_colsum[C_];
  __shared__ float s_wave[WAVES];

  const int tid = threadIdx.x;
  const int ln  = tid & 31;        // lane in wave32
  const int wv  = tid >> 5;
  const int b     = blockIdx.x / CHUNKS_PER_B;
  const int chunk = blockIdx.x % CHUNKS_PER_B;

  // colsum(trans)[k] = sum_i trans[i,k]  (for sum_g), once per block into LDS
  if (tid < C_) {
    float s = 0.f;
    #pragma unroll
    for (int i = 0; i < C_; ++i) s += trans[i * C_ + tid];
    s_colsum[tid] = s;
  }
  __syncthreads();

  const int pr  = ln & 15;   // which of the 16 pairs (= WMMA A-row M)
  const int hi  = ln >> 4;   // 0: j%4 in {0,1}; 1: j%4 in {2,3}  (A/B K halves)
  const int kof = 2 * hi;

  // B matrices (exp(trans)) in WMMA 4x16 layout, per K-step s (K = 4s..4s+3):
  //   lanes 0-15: (K=4s+kof, N=lane)  lanes 16-31: (K=4s+kof, N=lane-16)
  //   B[j][i] = exp(trans[i][j]);  bm0: i=N, bm1: i=N+16.
  v2f bm0[8], bm1[8];
  #pragma unroll
  for (int s = 0; s < 8; ++s) {
    const int j0 = 4 * s + kof;
    bm0[s].x = __expf(trans[pr * C_ + j0]);
    bm0[s].y = __expf(trans[pr * C_ + j0 + 1]);
    bm1[s].x = __expf(trans[(pr + 16) * C_ + j0]);
    bm1[s].y = __expf(trans[(pr + 16) * C_ + j0 + 1]);
  }

  const int len_b = lens[b];
  float acc     = 0.f;   // direct (already-scaled) contributions
  float acc_l2  = 0.f;   // accumulated log2(S) terms; scaled by 32*ln2 at end

  for (int it = 0; it < ITERS; ++it) {
    const int t0 = chunk * CHUNK_T + wv * (ITERS * 16) + it * 16;
    const int t  = t0 + pr;                           // this lane's pair time
    const size_t rowbase = ((size_t)b * T_ + t) * C_;

    // ---- cheap per-pair terms: sum_h, sum_g (lanes 0-15 own the pairs) ----
    if (hi == 0) {
      const int lab = tt[(size_t)b * T_ + t];
      if (lab < C_) {
        acc -= tp[rowbase + lab];                     // sum_h
        if (t < T_ - 1) {
          const int lab2 = tt[(size_t)b * T_ + t + 1];
          if (lab2 < C_) acc -= s_colsum[lab];        // sum_g
        }
      }
    }

    const bool validZ = (t >= 1) && (t <= len_b);
    const unsigned maskZ = (unsigned)(__ballot(validZ) & 0xFFFFu);
    const bool anyT0 = (t0 == 0);                     // wave-uniform

    if (maskZ || anyT0) {
      // Load this lane's 16 e-values: j = 4s+kof, 4s+kof+1 (row covered by
      // lane pair L / L+16). Track max on the fly.
      float ex[16];
      float mymax = -3.4e38f;
      #pragma unroll
      for (int s = 0; s < 8; ++s) {
        const float2 v = *(const float2*)(tp + rowbase + 4 * s + kof);
        ex[2 * s]     = v.x;
        ex[2 * s + 1] = v.y;
        mymax = fmaxf(mymax, fmaxf(v.x, v.y));
      }
      const float m = fmaxf(mymax, __shfl_xor(mymax, 16, 32));

      if (anyT0) {                                    // only the t0==0 tile
        float mysum = 0.f;
        #pragma unroll
        for (int q = 0; q < 16; ++q) mysum += ex[q];
        // lanes 0 and 16 each hold half of pair 0's row sum; both add;
        // the final cross-lane block reduction completes C * sum_j tp[b,0,j].
        if (t == 0) acc += 32.f * mysum;
      }

      if (maskZ) {
        // A (16x4 f32) per K-step: VGPR0/1 hold K=4s+kof / +1 for M=pr.
        v2f a[8];
        #pragma unroll
        for (int s = 0; s < 8; ++s) {
          a[s].x = __expf(ex[2 * s]     - m);
          a[s].y = __expf(ex[2 * s + 1] - m);
        }
        v8f c0 = {}; v8f c1 = {};
        #pragma unroll
        for (int s = 0; s < 8; ++s) {
          c0 = __builtin_amdgcn_wmma_f32_16x16x4_f32(false, a[s], false, bm0[s],
                                                     (short)0, c0, false, false);
          c1 = __builtin_amdgcn_wmma_f32_16x16x4_f32(false, a[s], false, bm1[s],
                                                     (short)0, c1, false, false);
        }
        // D layout: VGPR r: lanes0-15 -> (M=r, N=lane); lanes16-31 -> (M=r+8).
        // S_i >= 1 always (p_max=1, exp(trans)>=1) => raw v_log is safe.
        if (maskZ == 0xFFFFu) {                       // common: all 16 valid
          #pragma unroll
          for (int r = 0; r < 8; ++r)
            acc_l2 += RAW_LOG2(c0[r]) + RAW_LOG2(c1[r]);
        } else {                                      // boundary tile
          #pragma unroll
          for (int r = 0; r < 8; ++r) {
            const int   sel = r + 8 * hi;
            const float w   = (float)((maskZ >> sel) & 1u);
            acc_l2 += w * (RAW_LOG2(c0[r]) + RAW_LOG2(c1[r]));
          }
        }
        if (validZ && hi == 0) acc += 1024.f * m;     // C * 32 * m
      }
    }
  }

  acc += (32.f * LN2_F) * acc_l2;                     // C * sum log S (nat log)

  // block reduction (wave shuffle + LDS), fixed order -> deterministic
  acc += __shfl_xor(acc, 1, 32);
  acc += __shfl_xor(acc, 2, 32);
  acc += __shfl_xor(acc, 4, 32);
  acc += __shfl_xor(acc, 8, 32);
  acc += __shfl_xor(acc, 16, 32);
  if (ln == 0) s_wave[wv] = acc;
  __syncthreads();
  if (tid == 0) {
    float s = 0.f;
    #pragma unroll
    for (int w = 0; w < WAVES; ++w) s += s_wave[w];
    partials[blockIdx.x] = s;
  }
}

__global__ __launch_bounds__(256)
void crf_reduce(const float* __restrict__ partials, float* __restrict__ out, int n)
{
  __shared__ float sm[256];
  float s = 0.f;
  for (int i = threadIdx.x; i < n; i += 256) s += partials[i]; // fixed order
  sm[threadIdx.x] = s;
  __syncthreads();
  for (int st = 128; st > 0; st >>= 1) {
    if ((int)threadIdx.x < st) sm[threadIdx.x] += sm[threadIdx.x + st];
    __syncthreads();
  }
  if (threadIdx.x == 0) out[0] = sm[0] / (float)B_;   // mean over B
}

extern "C" void kernel_launch(void* const* d_in, const int* in_sizes, int n_in,
                              void* d_out, int out_size, void* d_ws, size_t ws_size,
                              hipStream_t stream)
{
  (void)in_sizes; (void)n_in; (void)out_size; (void)ws_size;
  const float* tp    = (const float*)d_in[0];
  const float* trans = (const float*)d_in[1];
  const int*   tt    = (const int*)d_in[2];
  const int*   lens  = (const int*)d_in[3];
  float* partials = (float*)d_ws;            // B_*CHUNKS_PER_B floats = 8 KB
  float* out      = (float*)d_out;

  const int nblocks = B_ * CHUNKS_PER_B;     // 2048
  crf_main<<<nblocks, 256, 0, stream>>>(tp, trans, tt, lens, partials);
  crf_reduce<<<1, 256, 0, stream>>>(partials, out, nblocks);
}